// DeformConv2d_3246995276085
// MI455X (gfx1250) — compile-verified
//
#include <hip/hip_runtime.h>
#include <hip/hip_bf16.h>
#include <math.h>
#include <stdint.h>

typedef __attribute__((ext_vector_type(2))) float v2f;
typedef __attribute__((ext_vector_type(8))) float v8f;

#define BN 4
#define CC 64
#define OO 64
#define HH 96
#define WW 96
#define NDIR 9
#define HWSZ (HH * WW)               // 9216
#define MTOT (BN * HWSZ)             // 36864
#define K9   (9 * CC)                // 576
#define LP   66                      // LDS row pitch (floats), 8B aligned, conflict-free

// CDNA5 async global->LDS copy: 16 bytes per lane, ASYNCcnt-tracked, no VGPR
// round-trip. INST_OFFSET is added to BOTH the LDS and global addresses.
#define ASYNC_LDS_CP16(ldsoff, gaddr, OFFS)                                  \
    asm volatile("global_load_async_to_lds_b128 %0, %1, off offset:" #OFFS   \
                 :: "v"(ldsoff), "v"(gaddr) : "memory")
#define ASYNC_CP_ROW128(ldsoff, gaddr)                                       \
    do {                                                                     \
        ASYNC_LDS_CP16(ldsoff, gaddr, 0);                                    \
        ASYNC_LDS_CP16(ldsoff, gaddr, 16);                                   \
        ASYNC_LDS_CP16(ldsoff, gaddr, 32);                                   \
        ASYNC_LDS_CP16(ldsoff, gaddr, 48);                                   \
        ASYNC_LDS_CP16(ldsoff, gaddr, 64);                                   \
        ASYNC_LDS_CP16(ldsoff, gaddr, 80);                                   \
        ASYNC_LDS_CP16(ldsoff, gaddr, 96);                                   \
        ASYNC_LDS_CP16(ldsoff, gaddr, 112);                                  \
    } while (0)
#define ASYNC_WAIT0() asm volatile("s_wait_asynccnt 0x0" ::: "memory")

// ---------------------------------------------------------------------------
// 1) NCHW -> NHWC transpose of x
// ---------------------------------------------------------------------------
__global__ void transpose_kernel(const float* __restrict__ x, float* __restrict__ xt) {
    int idx = blockIdx.x * blockDim.x + threadIdx.x;
    if (idx >= BN * CC * HWSZ) return;
    int w = idx % WW;
    int t = idx / WW;
    int h = t % HH; t /= HH;
    int c = t % CC;
    int b = t / CC;
    xt[((b * HH + h) * WW + w) * CC + c] = x[idx];
}

// ---------------------------------------------------------------------------
// 2) Pack weights into [N][k = tap*64 + c] GEMM-B layout
//    Bt1: 48 cols = w_off1(18) | w_off2(18) | w_mod(9) | zero-pad(3)
//    Bt3: 64 cols = conv_weight
// ---------------------------------------------------------------------------
__global__ void pack_kernel(const float* __restrict__ w1, const float* __restrict__ w2,
                            const float* __restrict__ wm, const float* __restrict__ wc,
                            float* __restrict__ Bt1, float* __restrict__ Bt3) {
    int idx = blockIdx.x * blockDim.x + threadIdx.x;
    if (idx < 48 * K9) {
        int n = idx / K9, k = idx % K9;
        int tap = k >> 6, c = k & 63;
        float v = 0.0f;
        if (n < 18)      v = w1[(n * CC + c) * 9 + tap];
        else if (n < 36) v = w2[((n - 18) * CC + c) * 9 + tap];
        else if (n < 45) v = wm[((n - 36) * CC + c) * 9 + tap];
        Bt1[idx] = v;
    } else if (idx < 48 * K9 + 64 * K9) {
        int j = idx - 48 * K9;
        int n = j / K9, k = j % K9;
        int tap = k >> 6, c = k & 63;
        Bt3[j] = wc[(n * CC + c) * 9 + tap];
    }
}

// ---------------------------------------------------------------------------
// 3) Offset/modulation conv as implicit GEMM (M=36864, N=48, K=576) with
//    V_WMMA_F32_16X16X4_F32; per-tap tiles staged with async global->LDS.
//    Epilogue builds sampling locations (ix,iy) + sigmoid modulation.
// ---------------------------------------------------------------------------
__global__ __launch_bounds__(128) void conv_off_kernel(
    const float* __restrict__ xt, const float* __restrict__ Bt,
    const float* __restrict__ b1, const float* __restrict__ b2,
    const float* __restrict__ bm, const float* __restrict__ alpha_p,
    float* __restrict__ loc, float* __restrict__ mod) {
    __shared__ float As[64 * LP];
    __shared__ float Bs[48 * LP];
    __shared__ float Ds[64 * 48];

    const int m0   = blockIdx.x * 64;
    const int t    = threadIdx.x;
    const int lane = t & 31;
    const int wave = t >> 5;
    const int lo   = lane & 15;
    const int hi   = lane >> 4;

    v8f c0 = {}, c1 = {}, c2 = {};

    for (int tap = 0; tap < 9; ++tap) {
        const int kh = tap / 3, kw = tap % 3;
        __syncthreads();
        {   // A tile: 64 rows x 64 ch; 2 threads/row, 128B each via async copies
            int r = t >> 1, half = t & 1;
            int m = m0 + r;
            int b = m / HWSZ; int rem = m - b * HWSZ;
            int h = rem / WW; int w = rem - h * WW;
            int hh = h + kh - 1, ww = w + kw - 1;
            unsigned ldsoff = (unsigned)(uintptr_t)&As[r * LP + half * 32];
            if (hh >= 0 && hh < HH && ww >= 0 && ww < WW) {
                unsigned long long ga = (unsigned long long)(uintptr_t)
                    (xt + ((size_t)(b * HH + hh) * WW + ww) * CC + half * 32);
                ASYNC_CP_ROW128(ldsoff, ga);
            } else {
                float4 z = {0.f, 0.f, 0.f, 0.f};
                float4* d4 = (float4*)&As[r * LP + half * 32];
                #pragma unroll
                for (int j = 0; j < 8; ++j) d4[j] = z;
            }
        }
        if (t < 96) {  // B tile: 48 rows x 64 k via async copies
            int n = t >> 1, half = t & 1;
            unsigned ldsoff = (unsigned)(uintptr_t)&Bs[n * LP + half * 32];
            unsigned long long gb = (unsigned long long)(uintptr_t)
                (Bt + (size_t)n * K9 + tap * CC + half * 32);
            ASYNC_CP_ROW128(ldsoff, gb);
        }
        ASYNC_WAIT0();
        __syncthreads();

        const float* arow = &As[(wave * 16 + lo) * LP + hi * 2];
        const float* br0  = &Bs[(lo)      * LP + hi * 2];
        const float* br1  = &Bs[(16 + lo) * LP + hi * 2];
        const float* br2  = &Bs[(32 + lo) * LP + hi * 2];
        #pragma unroll
        for (int ks = 0; ks < 16; ++ks) {
            v2f a   = *(const v2f*)(arow + ks * 4);
            v2f bb0 = *(const v2f*)(br0 + ks * 4);
            v2f bb1 = *(const v2f*)(br1 + ks * 4);
            v2f bb2 = *(const v2f*)(br2 + ks * 4);
            c0 = __builtin_amdgcn_wmma_f32_16x16x4_f32(false, a, false, bb0, (short)0, c0, false, false);
            c1 = __builtin_amdgcn_wmma_f32_16x16x4_f32(false, a, false, bb1, (short)0, c1, false, false);
            c2 = __builtin_amdgcn_wmma_f32_16x16x4_f32(false, a, false, bb2, (short)0, c2, false, false);
        }
    }

    __syncthreads();
    #pragma unroll
    for (int i = 0; i < 8; ++i) {
        int r = wave * 16 + hi * 8 + i;
        Ds[r * 48 + lo]      = c0[i];
        Ds[r * 48 + 16 + lo] = c1[i];
        Ds[r * 48 + 32 + lo] = c2[i];
    }
    __syncthreads();

    if (lane < 16) {
        int r = wave * 16 + lane;
        int m = m0 + r;
        int b = m / HWSZ; int rem = m - b * HWSZ;
        int h = rem / WW; int w = rem - h * WW;
        (void)b;
        float alpha = alpha_p[0];
        const float dirx[9] = {0.f,0.f,0.f,1.f,1.f,1.f,-1.f,-1.f,-1.f};
        const float diry[9] = {0.f,1.f,-1.f,0.f,1.f,-1.f,0.f,1.f,-1.f};
        const float* v = &Ds[r * 48];
        #pragma unroll
        for (int d = 0; d < NDIR; ++d) {
            float o1x = v[d]             + b1[d];
            float o1y = v[NDIR + d]      + b1[NDIR + d];
            float o2x = v[18 + d]        + b2[d];
            float o2y = v[18 + NDIR + d] + b2[NDIR + d];
            float md  = v[36 + d]        + bm[d];
            md = 1.0f / (1.0f + expf(-md));
            float gx = (float)h + dirx[d] + alpha * o1x + (1.0f - alpha) * o2x;
            float gy = (float)w + diry[d] + alpha * o1y + (1.0f - alpha) * o2y;
            float ixf = ((gx + 1.0f) * (float)WW - 1.0f) * 0.5f;
            float iyf = ((gy + 1.0f) * (float)HH - 1.0f) * 0.5f;
            float2 lv; lv.x = ixf; lv.y = iyf;
            *(float2*)(loc + (size_t)(m * NDIR + d) * 2) = lv;
            mod[m * NDIR + d] = md;
        }
    }
}

// ---------------------------------------------------------------------------
// 4) Deformable bilinear sampling + modulation sum.
//    One wave per pixel; 2 channels per lane; NHWC gathers are coalesced.
// ---------------------------------------------------------------------------
__global__ __launch_bounds__(256) void sample_kernel(
    const float* __restrict__ xt, const float* __restrict__ loc,
    const float* __restrict__ mod, float* __restrict__ feat) {
    int m = blockIdx.x * (blockDim.x >> 5) + (threadIdx.x >> 5);
    if (m >= MTOT) return;
    int lane = threadIdx.x & 31;
    int b = m / HWSZ;
    int c0i = lane * 2;
    const float* src = xt + (size_t)b * HWSZ * CC;

    float ax = 0.f, ay = 0.f;
    for (int d = 0; d < NDIR; ++d) {
        float2 lv = *(const float2*)(loc + (size_t)(m * NDIR + d) * 2);
        float md = mod[m * NDIR + d];
        float ixf = lv.x, iyf = lv.y;
        float x0f = floorf(ixf), y0f = floorf(iyf);
        float wx1 = ixf - x0f, wy1 = iyf - y0f;
        float wx0 = 1.f - wx1, wy0 = 1.f - wy1;
        int ix0 = (int)x0f, iy0 = (int)y0f;
        int ix1 = ix0 + 1,  iy1 = iy0 + 1;

        float sx = 0.f, sy = 0.f;
        #pragma unroll
        for (int q = 0; q < 4; ++q) {
            int yi = (q < 2) ? iy0 : iy1;
            int xi = (q & 1) ? ix1 : ix0;
            float cw = ((q < 2) ? wy0 : wy1) * ((q & 1) ? wx1 : wx0);
            float valid = (yi >= 0 && yi < HH && xi >= 0 && xi < WW) ? 1.f : 0.f;
            int yc = min(max(yi, 0), HH - 1);
            int xc = min(max(xi, 0), WW - 1);
            float2 v = *(const float2*)(src + ((size_t)yc * WW + xc) * CC + c0i);
            float f = cw * valid;
            sx += f * v.x;
            sy += f * v.y;
        }
        ax += md * sx;
        ay += md * sy;
    }
    float2 o; o.x = ax; o.y = ay;
    *(float2*)(feat + (size_t)m * CC + c0i) = o;
}

// ---------------------------------------------------------------------------
// 5) Final conv as implicit GEMM (M=36864, N=64, K=576) with f32 WMMA and
//    async global->LDS staging. Epilogue scatters fp32 NCHW output.
// ---------------------------------------------------------------------------
__global__ __launch_bounds__(128) void conv_out_kernel(
    const float* __restrict__ feat, const float* __restrict__ Bt,
    float* __restrict__ out) {
    __shared__ float As[64 * LP];
    __shared__ float Bs[64 * LP];

    const int m0   = blockIdx.x * 64;
    const int t    = threadIdx.x;
    const int lane = t & 31;
    const int wave = t >> 5;
    const int lo   = lane & 15;
    const int hi   = lane >> 4;

    v8f c0 = {}, c1 = {}, c2 = {}, c3 = {};

    for (int tap = 0; tap < 9; ++tap) {
        const int kh = tap / 3, kw = tap % 3;
        __syncthreads();
        {
            int r = t >> 1, half = t & 1;
            int m = m0 + r;
            int b = m / HWSZ; int rem = m - b * HWSZ;
            int h = rem / WW; int w = rem - h * WW;
            int hh = h + kh - 1, ww = w + kw - 1;
            unsigned ldsoff = (unsigned)(uintptr_t)&As[r * LP + half * 32];
            if (hh >= 0 && hh < HH && ww >= 0 && ww < WW) {
                unsigned long long ga = (unsigned long long)(uintptr_t)
                    (feat + ((size_t)(b * HH + hh) * WW + ww) * CC + half * 32);
                ASYNC_CP_ROW128(ldsoff, ga);
            } else {
                float4 z = {0.f, 0.f, 0.f, 0.f};
                float4* d4 = (float4*)&As[r * LP + half * 32];
                #pragma unroll
                for (int j = 0; j < 8; ++j) d4[j] = z;
            }
        }
        {
            int n = t >> 1, half = t & 1;
            unsigned ldsoff = (unsigned)(uintptr_t)&Bs[n * LP + half * 32];
            unsigned long long gb = (unsigned long long)(uintptr_t)
                (Bt + (size_t)n * K9 + tap * CC + half * 32);
            ASYNC_CP_ROW128(ldsoff, gb);
        }
        ASYNC_WAIT0();
        __syncthreads();

        const float* arow = &As[(wave * 16 + lo) * LP + hi * 2];
        const float* br0  = &Bs[(lo)      * LP + hi * 2];
        const float* br1  = &Bs[(16 + lo) * LP + hi * 2];
        const float* br2  = &Bs[(32 + lo) * LP + hi * 2];
        const float* br3  = &Bs[(48 + lo) * LP + hi * 2];
        #pragma unroll
        for (int ks = 0; ks < 16; ++ks) {
            v2f a   = *(const v2f*)(arow + ks * 4);
            v2f bb0 = *(const v2f*)(br0 + ks * 4);
            v2f bb1 = *(const v2f*)(br1 + ks * 4);
            v2f bb2 = *(const v2f*)(br2 + ks * 4);
            v2f bb3 = *(const v2f*)(br3 + ks * 4);
            c0 = __builtin_amdgcn_wmma_f32_16x16x4_f32(false, a, false, bb0, (short)0, c0, false, false);
            c1 = __builtin_amdgcn_wmma_f32_16x16x4_f32(false, a, false, bb1, (short)0, c1, false, false);
            c2 = __builtin_amdgcn_wmma_f32_16x16x4_f32(false, a, false, bb2, (short)0, c2, false, false);
            c3 = __builtin_amdgcn_wmma_f32_16x16x4_f32(false, a, false, bb3, (short)0, c3, false, false);
        }
    }

    // Epilogue: D[M=i+8*hi][N=tt*16+lo] -> out NCHW
    #pragma unroll
    for (int i = 0; i < 8; ++i) {
        int r = wave * 16 + hi * 8 + i;
        int m = m0 + r;
        int b = m / HWSZ; int rem = m - b * HWSZ;
        int h = rem / WW; int w = rem - h * WW;
        out[(((size_t)b * OO + (0 * 16 + lo)) * HH + h) * WW + w] = c0[i];
        out[(((size_t)b * OO + (1 * 16 + lo)) * HH + h) * WW + w] = c1[i];
        out[(((size_t)b * OO + (2 * 16 + lo)) * HH + h) * WW + w] = c2[i];
        out[(((size_t)b * OO + (3 * 16 + lo)) * HH + h) * WW + w] = c3[i];
    }
}

// ---------------------------------------------------------------------------
extern "C" void kernel_launch(void* const* d_in, const int* in_sizes, int n_in,
                              void* d_out, int out_size, void* d_ws, size_t ws_size,
                              hipStream_t stream) {
    const float* x      = (const float*)d_in[0];
    const float* w_off1 = (const float*)d_in[1];
    const float* b_off1 = (const float*)d_in[2];
    const float* w_off2 = (const float*)d_in[3];
    const float* b_off2 = (const float*)d_in[4];
    const float* w_mod  = (const float*)d_in[5];
    const float* b_mod  = (const float*)d_in[6];
    const float* conv_w = (const float*)d_in[7];
    const float* alpha  = (const float*)d_in[8];
    float* out = (float*)d_out;

    float* ws   = (float*)d_ws;
    float* xt   = ws;                                   // MTOT*CC        = 2359296
    float* Bt1  = xt   + (size_t)MTOT * CC;             // 48*K9          = 27648
    float* Bt3  = Bt1  + (size_t)48 * K9;               // 64*K9          = 36864
    float* locw = Bt3  + (size_t)64 * K9;               // MTOT*NDIR*2    = 663552
    float* modw = locw + (size_t)MTOT * NDIR * 2;       // MTOT*NDIR      = 331776
    float* feat = modw + (size_t)MTOT * NDIR;           // MTOT*CC        = 2359296

    transpose_kernel<<<(BN * CC * HWSZ + 255) / 256, 256, 0, stream>>>(x, xt);
    pack_kernel<<<((48 + 64) * K9 + 255) / 256, 256, 0, stream>>>(w_off1, w_off2, w_mod, conv_w, Bt1, Bt3);
    conv_off_kernel<<<MTOT / 64, 128, 0, stream>>>(xt, Bt1, b_off1, b_off2, b_mod, alpha, locw, modw);
    sample_kernel<<<MTOT / 8, 256, 0, stream>>>(xt, locw, modw, feat);
    conv_out_kernel<<<MTOT / 64, 128, 0, stream>>>(feat, Bt3, out);
}